// L2BoundedLTICell_45732811768499
// MI455X (gfx1250) — compile-verified
//
#include <hip/hip_runtime.h>
#include <math.h>

// L2BoundedLTICell for MI455X (gfx1250, wave32, WMMA).
// Pipeline: power-iteration sigma -> scale/split K -> Newton-Schulz S^-1 (WMMA GEMMs)
//   -> A,B,C,D via WMMA GEMMs -> W=U@B^T and UD=U@D^T (parallel WMMA GEMMs)
//   -> persistent 32-block cooperative scan (A slices in LDS, fp32 WMMA, atomic grid barrier)
//   -> output += PRE@C^T (WMMA GEMM with layout-permuting epilogue) -> states transpose copy.

typedef __attribute__((ext_vector_type(2))) float v2f;
typedef __attribute__((ext_vector_type(8))) float v8f;

#define WMMA_F32(a, b, c) \
  __builtin_amdgcn_wmma_f32_16x16x4_f32(false, (a), false, (b), (short)0, (c), false, false)

// ---------------------------------------------------------------------------
// Small utility kernels
// ---------------------------------------------------------------------------
__global__ void set_const_k(float* __restrict__ p, float v, int n) {
  int i = blockIdx.x * 256 + threadIdx.x;
  if (i < n) p[i] = v;
}

__global__ void set_identity_k(float* __restrict__ X, int n) {
  int idx = blockIdx.x * 256 + threadIdx.x;
  if (idx >= n * n) return;
  int i = idx / n, j = idx % n;
  X[idx] = (i == j) ? 1.0f : 0.0f;
}

__global__ void two_i_minus_k(float* __restrict__ E, int n) {
  int idx = blockIdx.x * 256 + threadIdx.x;
  if (idx >= n * n) return;
  int i = idx / n, j = idx % n;
  E[idx] = ((i == j) ? 2.0f : 0.0f) - E[idx];
}

// y[r] = dot(M[r,:], x) : one block per row
__global__ void mv_row_k(const float* __restrict__ M, const float* __restrict__ x,
                         float* __restrict__ y, int n) {
  __shared__ float red[256];
  int r = blockIdx.x;
  float s = 0.0f;
  for (int c = threadIdx.x; c < n; c += 256) s += M[(size_t)r * n + c] * x[c];
  red[threadIdx.x] = s;
  __syncthreads();
  for (int o = 128; o > 0; o >>= 1) {
    if (threadIdx.x < o) red[threadIdx.x] += red[threadIdx.x + o];
    __syncthreads();
  }
  if (threadIdx.x == 0) y[r] = red[0];
}

// y[c] = dot(M[:,c], x) : one thread per column (lane-coalesced column walk)
__global__ void mv_col_k(const float* __restrict__ M, const float* __restrict__ x,
                         float* __restrict__ y, int n) {
  int c = blockIdx.x * 256 + threadIdx.x;
  if (c >= n) return;
  float s = 0.0f;
  for (int r = 0; r < n; ++r) s += M[(size_t)r * n + c] * x[r];
  y[c] = s;
}

// norm of src; optional dst = src/norm ; optional *sigma_out = norm
__global__ void normalize_k(const float* __restrict__ src, float* __restrict__ dst,
                            float* __restrict__ sigma_out, int n) {
  __shared__ float red[256];
  float s = 0.0f;
  for (int i = threadIdx.x; i < n; i += 256) { float v = src[i]; s += v * v; }
  red[threadIdx.x] = s;
  __syncthreads();
  for (int o = 128; o > 0; o >>= 1) {
    if (threadIdx.x < o) red[threadIdx.x] += red[threadIdx.x + o];
    __syncthreads();
  }
  float nrm = sqrtf(red[0]);
  if (sigma_out && threadIdx.x == 0) *sigma_out = nrm;
  if (dst) {
    float inv = 1.0f / fmaxf(nrm, 1e-30f);
    for (int i = threadIdx.x; i < n; i += 256) dst[i] = src[i] * inv;
  }
}

// scal[0]=sigma -> scal[1]=1/(max(sigma,1e-5)+eps), scal[2]=exp(log_gamma)
__global__ void prep_scalars_k(float* __restrict__ scal, const float* __restrict__ lg) {
  if (threadIdx.x == 0) {
    float s = fmaxf(scal[0], 1e-5f);
    scal[1] = 1.0f / (s + 0.002f);
    scal[2] = expf(lg[0]);
  }
}

// split 1024x1024 K_raw*inv_sigma into K11,K12,K21, K22*gamma (512x512 each)
__global__ void split_k(const float* __restrict__ Kr, float* __restrict__ K11,
                        float* __restrict__ K12, float* __restrict__ K21,
                        float* __restrict__ K22d, const float* __restrict__ scal) {
  int idx = blockIdx.x * 256 + threadIdx.x;
  if (idx >= 1024 * 1024) return;
  int r = idx >> 10, c = idx & 1023;
  float inv_sigma = scal[1], gamma = scal[2];
  float v = Kr[idx] * inv_sigma;
  int o = (r & 511) * 512 + (c & 511);
  if (r < 512) { if (c < 512) K11[o] = v; else K12[o] = v; }
  else         { if (c < 512) K21[o] = v; else K22d[o] = v * gamma; }
}

// states[b][t][d] = Xbuf[t][b][d]
__global__ void copy_states_k(const float* __restrict__ X, float* __restrict__ out,
                              int T, size_t total) {
  size_t idx = (size_t)blockIdx.x * 256 + threadIdx.x;
  if (idx >= total) return;
  int d = (int)(idx & 511);
  size_t r = idx >> 9;               // r = b*(T+1) + t
  int t = (int)(r % (size_t)(T + 1));
  int b = (int)(r / (size_t)(T + 1));
  out[idx] = X[((size_t)t * 16 + b) * 512 + d];
}

// ---------------------------------------------------------------------------
// fp32 WMMA GEMM:  C = alpha * P(MxK,row) @ op(Q)   [TRANSQ: Q is NxK -> P@Q^T]
// 256 threads = 8 waves arranged 2(M)x4(N); each wave computes a 32x32 tile
// (4 accumulators, 2 A-frags + 2 B-frags per k-step -> 4 WMMAs per 4 loads).
// Block tile = 64(M) x 128(N).  Grid: (M/64, N/128).
// epi: 0 plain write; 1 remap row m=(b*T+t) -> (t*16+b) (time-major store);
//      2 remap row m=(t*16+b) -> (b*T+t) and accumulate (+=).
// ---------------------------------------------------------------------------
template <bool TRANSQ>
__global__ __launch_bounds__(256) void gemm_f32(
    const float* __restrict__ P, const float* __restrict__ Q, float* __restrict__ C,
    int M, int N, int K, const float* __restrict__ alpha_ptr, int epi, int T) {
  (void)M;
  const int tid  = threadIdx.x;
  const int wv   = tid >> 5, lane = tid & 31;
  const int half = lane >> 4, l16 = lane & 15;
  const int wm = wv >> 2, wn = wv & 3;
  const int m0 = blockIdx.x * 64 + wm * 32;
  const int n0 = blockIdx.y * 128 + wn * 32;
  const int mA = m0 + l16, nB = n0 + l16;

  v8f acc00 = {}, acc01 = {}, acc10 = {}, acc11 = {};
#pragma unroll 4
  for (int k = 0; k < K; k += 4) {
    const int kb = k + 2 * half;              // VGPR0 holds K=kb, VGPR1 holds K=kb+1
    v2f a0 = *(const v2f*)(P + (size_t)mA * K + kb);
    v2f a1 = *(const v2f*)(P + (size_t)(mA + 16) * K + kb);
    v2f b0, b1;
    if (TRANSQ) {
      b0 = *(const v2f*)(Q + (size_t)nB * K + kb);
      b1 = *(const v2f*)(Q + (size_t)(nB + 16) * K + kb);
    } else {
      b0.x = Q[(size_t)kb * N + nB];
      b0.y = Q[(size_t)(kb + 1) * N + nB];
      b1.x = Q[(size_t)kb * N + nB + 16];
      b1.y = Q[(size_t)(kb + 1) * N + nB + 16];
    }
    acc00 = WMMA_F32(a0, b0, acc00);
    acc01 = WMMA_F32(a0, b1, acc01);
    acc10 = WMMA_F32(a1, b0, acc10);
    acc11 = WMMA_F32(a1, b1, acc11);
  }

  const float alpha = alpha_ptr ? alpha_ptr[0] : 1.0f;
  auto store_tile = [&](v8f acc, int mt, int nt) {
#pragma unroll
    for (int r = 0; r < 8; ++r) {
      const int mC = mt + r + 8 * half;       // C layout: VGPR r, lanes>=16 -> M=r+8
      const int nC = nt + l16;
      float val = acc[r] * alpha;
      if (epi == 0) {
        C[(size_t)mC * N + nC] = val;
      } else if (epi == 1) {
        int b_ = mC / T, t_ = mC % T;
        C[((size_t)t_ * 16 + b_) * N + nC] = val;
      } else {
        int t_ = mC >> 4, b_ = mC & 15;
        C[((size_t)b_ * T + t_) * N + nC] += val;
      }
    }
  };
  store_tile(acc00, m0, n0);
  store_tile(acc01, m0, n0 + 16);
  store_tile(acc10, m0 + 16, n0);
  store_tile(acc11, m0 + 16, n0 + 16);
}

// ---------------------------------------------------------------------------
// Persistent cooperative scan: Xbuf[t+1] = Xbuf[t] @ A^T + W[t]
// 32 blocks; block j owns state columns [16j,16j+16); its 16x512 slice of A
// lives in LDS (row pitch 516 floats -> conflict-free b64 reads).
// Per step: 8 waves split K=512 (64 each, 16 WMMAs), LDS sum of 8 wave partials, grid barrier.
// ---------------------------------------------------------------------------
__global__ __launch_bounds__(256) void scan_kernel(
    const float* __restrict__ Amat, const float* __restrict__ Wbuf,
    const float* __restrict__ state0, float* __restrict__ Xbuf,
    unsigned* __restrict__ cnt, int Tsteps, int nblocks) {
  __shared__ float Alds[16 * 516];
  __shared__ float pbuf[8 * 256];
  const int tid  = threadIdx.x;
  const int wv   = tid >> 5, lane = tid & 31;
  const int half = lane >> 4, l16 = lane & 15;
  const int j0 = blockIdx.x * 16;

  for (int i = tid; i < 16 * 512; i += 256) {
    int j = i >> 9, k = i & 511;
    Alds[j * 516 + k] = Amat[(size_t)(j0 + j) * 512 + k];
  }
  {  // x0 slice from input state
    int b = tid >> 4, d = tid & 15;
    Xbuf[(size_t)b * 512 + j0 + d] = state0[(size_t)b * 512 + j0 + d];
  }
  __threadfence();
  unsigned bstep = 1;
  if (tid == 0) {
    __hip_atomic_fetch_add(cnt, 1u, __ATOMIC_ACQ_REL, __HIP_MEMORY_SCOPE_AGENT);
    while (__hip_atomic_load(cnt, __ATOMIC_ACQUIRE, __HIP_MEMORY_SCOPE_AGENT) <
           (unsigned)nblocks * bstep)
      __builtin_amdgcn_s_sleep(1);
  }
  __syncthreads();

  for (int t = 0; t < Tsteps; ++t) {
    const float* xrow = Xbuf + (size_t)t * 16 * 512;
    v8f acc = {};
    const int kbase = wv * 64;
#pragma unroll
    for (int i = 0; i < 16; ++i) {
      const int kb = kbase + i * 4 + 2 * half;
      v2f a = *(const v2f*)(xrow + (size_t)l16 * 512 + kb);     // x[b=l16][kb..kb+1]
      v2f b = *(const v2f*)(Alds + l16 * 516 + kb);             // A[j=l16][kb..kb+1]
      acc = WMMA_F32(a, b, acc);
    }
#pragma unroll
    for (int r = 0; r < 8; ++r) {
      int mrow = r + 8 * half;
      pbuf[wv * 256 + mrow * 16 + l16] = acc[r];
    }
    __syncthreads();
    {
      int b = tid >> 4, d = tid & 15;
      float s = Wbuf[((size_t)t * 16 + b) * 512 + j0 + d];
#pragma unroll
      for (int w = 0; w < 8; ++w) s += pbuf[w * 256 + tid];
      Xbuf[((size_t)(t + 1) * 16 + b) * 512 + j0 + d] = s;
    }
    __threadfence();
    ++bstep;
    if (tid == 0) {
      __hip_atomic_fetch_add(cnt, 1u, __ATOMIC_ACQ_REL, __HIP_MEMORY_SCOPE_AGENT);
      while (__hip_atomic_load(cnt, __ATOMIC_ACQUIRE, __HIP_MEMORY_SCOPE_AGENT) <
             (unsigned)nblocks * bstep)
        __builtin_amdgcn_s_sleep(1);
    }
    __syncthreads();
  }
}

// ---------------------------------------------------------------------------
extern "C" void kernel_launch(void* const* d_in, const int* in_sizes, int n_in,
                              void* d_out, int out_size, void* d_ws, size_t ws_size,
                              hipStream_t stream) {
  (void)in_sizes; (void)n_in; (void)out_size; (void)ws_size;
  const float* u      = (const float*)d_in[0];  // (16,2048,512)
  const float* state0 = (const float*)d_in[1];  // (16,512)
  const float* S      = (const float*)d_in[2];  // (512,512)
  const float* Kr     = (const float*)d_in[3];  // (1024,1024)
  const float* lg     = (const float*)d_in[4];  // scalar

  const int B = 16, T = 2048, DX = 512, N1 = 1024;
  float* out_output = (float*)d_out;                              // (16,2048,512)
  float* out_states = out_output + (size_t)B * T * DX;            // (16,2049,512)

  // workspace layout
  float* ws   = (float*)d_ws;
  float* scal = ws;                               // [0]=sigma [1]=inv_sigma [2]=gamma
  unsigned* cnt = (unsigned*)(ws + 8);
  float* v   = ws + 16;
  float* tv  = v + N1;
  float* vw  = tv + N1;
  const size_t MM = (size_t)DX * DX;
  float* K11  = vw + N1;
  float* K12  = K11 + MM;
  float* K21  = K12 + MM;
  float* K22d = K21 + MM;
  float* Xn   = K22d + MM;   // Newton-Schulz X
  float* En   = Xn + MM;
  float* Tn   = En + MM;
  float* Am   = Tn + MM;
  float* Bmm  = Am + MM;
  float* Cmm  = Bmm + MM;
  float* T2   = Cmm + MM;
  float* Wbuf = T2 + MM;                          // (T*16, 512) time-major
  float* Xbuf = Wbuf + (size_t)B * T * DX;        // (T+1, 16, 512) time-major

  // --- top singular value of K_raw via power iteration on K^T K ---
  set_const_k<<<(N1 + 255) / 256, 256, 0, stream>>>(v, 1.0f / 32.0f, N1);
  for (int it = 0; it < 96; ++it) {
    mv_row_k<<<N1, 256, 0, stream>>>(Kr, v, tv, N1);          // tv = K v
    mv_col_k<<<N1 / 256, 256, 0, stream>>>(Kr, tv, vw, N1);   // vw = K^T tv
    normalize_k<<<1, 256, 0, stream>>>(vw, v, nullptr, N1);
  }
  mv_row_k<<<N1, 256, 0, stream>>>(Kr, v, tv, N1);
  normalize_k<<<1, 256, 0, stream>>>(tv, nullptr, &scal[0], N1);  // sigma = ||K v||
  prep_scalars_k<<<1, 32, 0, stream>>>(scal, lg);
  split_k<<<(N1 * N1 + 255) / 256, 256, 0, stream>>>(Kr, K11, K12, K21, K22d, scal);

  // --- S^-1 via Newton-Schulz: X <- X (2I - S X), X0 = I ---
  set_identity_k<<<((int)MM + 255) / 256, 256, 0, stream>>>(Xn, DX);
  dim3 g512(DX / 64, DX / 128);
  for (int it = 0; it < 9; ++it) {
    gemm_f32<false><<<g512, 256, 0, stream>>>(S, Xn, En, DX, DX, DX, nullptr, 0, 0);
    two_i_minus_k<<<((int)MM + 255) / 256, 256, 0, stream>>>(En, DX);
    gemm_f32<false><<<g512, 256, 0, stream>>>(Xn, En, Tn, DX, DX, DX, nullptr, 0, 0);
    float* tmp = Xn; Xn = Tn; Tn = tmp;
  }

  // --- A = Sinv@K11@S ; Bm = gamma*(Sinv@K12) ; Cm = K21@S ---
  gemm_f32<false><<<g512, 256, 0, stream>>>(Xn, K11, T2, DX, DX, DX, nullptr, 0, 0);
  gemm_f32<false><<<g512, 256, 0, stream>>>(T2, S, Am, DX, DX, DX, nullptr, 0, 0);
  gemm_f32<false><<<g512, 256, 0, stream>>>(Xn, K12, Bmm, DX, DX, DX, &scal[2], 0, 0);
  gemm_f32<false><<<g512, 256, 0, stream>>>(K21, S, Cmm, DX, DX, DX, nullptr, 0, 0);

  // --- W = U @ Bm^T (time-major) ; UD = U @ D^T (straight into output) ---
  dim3 gBig(B * T / 64, DX / 128);
  gemm_f32<true><<<gBig, 256, 0, stream>>>(u, Bmm, Wbuf, B * T, DX, DX, nullptr, 1, T);
  gemm_f32<true><<<gBig, 256, 0, stream>>>(u, K22d, out_output, B * T, DX, DX, nullptr, 0, 0);

  // --- sequential scan (persistent cooperative kernel) ---
  hipMemsetAsync(cnt, 0, sizeof(unsigned), stream);
  scan_kernel<<<32, 256, 0, stream>>>(Am, Wbuf, state0, Xbuf, cnt, T, 32);

  // --- output += PRE @ C^T  (time-major rows -> batch-major accumulate) ---
  gemm_f32<true><<<gBig, 256, 0, stream>>>(Xbuf, Cmm, out_output, B * T, DX, DX, nullptr, 2, T);

  // --- states: (b,t,d) <- Xbuf(t,b,d), t in [0,2048] ---
  size_t tot = (size_t)B * (T + 1) * DX;
  copy_states_k<<<(unsigned)((tot + 255) / 256), 256, 0, stream>>>(Xbuf, out_states, T, tot);
}